// GNNStackStage_13073880449508
// MI455X (gfx1250) — compile-verified
//
#include <hip/hip_runtime.h>
#include <math.h>

// ---------------------------------------------------------------------------
// ResGatedGraphConv stack (L=3) for MI455X / gfx1250 (wave32, WMMA).
//   per layer:  k = x@Wk^T+bk ; q = x@Wq^T+bq ; v = x@Wv^T+bv   (WMMA f32)
//               agg = x@Ws^T                                    (WMMA f32)
//               agg += scatter_add( sigmoid(k[dst]+q[src])*v[src] )  (L2 atomics)
//               h = x + l2norm(relu(agg*gamma+beta))
//   final: out = l2norm(h)
// ---------------------------------------------------------------------------

typedef float v2f __attribute__((ext_vector_type(2)));
typedef float v8f __attribute__((ext_vector_type(8)));

#define N_NODES 40000
#define N_EDGES 640000
#define D       128
#define LYR     3
#define EPS_NORM 1e-12f

// ---------------------------------------------------------------------------
// Y[m][n] = sum_k X[m][k] * W[n][k]  (+ bias[n] if bias != nullptr)
// One block = 16 output rows; 8 waves each own a 16-wide column tile (D=128).
// Uses V_WMMA_F32_16X16X4_F32, chaining 32 WMMAs over K=128.
//
// ISA layouts (CDNA5 ISA):
//   A 16x4 f32 : lane l<16 -> M=l, {K=0,K=1}; lane 16+l -> M=l, {K=2,K=3}
//   B 4x16 f32 : lane l<16 -> N=l, {K=0,K=1}; lane 16+l -> N=l, {K=2,K=3}
//   C/D 16x16  : VGPR r -> M = r + 8*(lane>=16), N = lane&15
// ---------------------------------------------------------------------------
__global__ __launch_bounds__(256)
void gemm_xt_wmma(const float* __restrict__ X, const float* __restrict__ W,
                  const float* __restrict__ bias, float* __restrict__ Y)
{
    const int lane = threadIdx.x & 31;
    const int wave = threadIdx.x >> 5;      // 0..7 : column tile
    const int l15  = lane & 15;
    const int hi   = lane >> 4;             // 0 or 1 (K half)

    const int row0 = blockIdx.x * 16;       // 40000 % 16 == 0 -> no tail
    const int col0 = wave * 16;

    const float* __restrict__ arow = X + (size_t)(row0 + l15) * D;  // A row (M)
    const float* __restrict__ brow = W + (size_t)(col0 + l15) * D;  // B col (N) = W row

    v8f c = {};
    #pragma unroll
    for (int kk = 0; kk < D; kk += 4) {
        v2f a, b;
        a.x = arow[kk + 2 * hi];
        a.y = arow[kk + 2 * hi + 1];
        b.x = brow[kk + 2 * hi];
        b.y = brow[kk + 2 * hi + 1];
        // 8 args: (neg_a, A, neg_b, B, c_mod, C, reuse_a, reuse_b)
        c = __builtin_amdgcn_wmma_f32_16x16x4_f32(false, a, false, b,
                                                  (short)0, c, false, false);
    }

    const int   ncol = col0 + l15;
    const float bn   = bias ? bias[ncol] : 0.0f;
    #pragma unroll
    for (int r = 0; r < 8; ++r) {
        const int m = row0 + r + 8 * hi;
        Y[(size_t)m * D + ncol] = c[r] + bn;
    }
}

// ---------------------------------------------------------------------------
// One wave per edge: msg = sigmoid(k[dst]+q[src]) * v[src], atomically
// accumulated into agg[dst]. 128 dims / 32 lanes = float4 per lane.
// agg (20.5 MB) and k/q/v rows are L2-resident on the 192 MB L2.
// ---------------------------------------------------------------------------
__global__ __launch_bounds__(256)
void edge_msg(const int* __restrict__ src, const int* __restrict__ dst,
              const float* __restrict__ K, const float* __restrict__ Q,
              const float* __restrict__ V, float* __restrict__ agg)
{
    const int lane = threadIdx.x & 31;
    const int wave = threadIdx.x >> 5;
    const int e    = blockIdx.x * 8 + wave;     // 640000 % 8 == 0

    const int s = src[e];
    const int t = dst[e];
    const int co = lane * 4;

    const float4 kd = *(const float4*)(K + (size_t)t * D + co);
    const float4 qs = *(const float4*)(Q + (size_t)s * D + co);
    const float4 vs = *(const float4*)(V + (size_t)s * D + co);

    float4 m;
    m.x = vs.x / (1.0f + __expf(-(kd.x + qs.x)));
    m.y = vs.y / (1.0f + __expf(-(kd.y + qs.y)));
    m.z = vs.z / (1.0f + __expf(-(kd.z + qs.z)));
    m.w = vs.w / (1.0f + __expf(-(kd.w + qs.w)));

    float* ap = agg + (size_t)t * D + co;
    atomicAdd(ap + 0, m.x);
    atomicAdd(ap + 1, m.y);
    atomicAdd(ap + 2, m.z);
    atomicAdd(ap + 3, m.w);
}

// ---------------------------------------------------------------------------
// One wave per node: BN(eval) -> ReLU -> row L2-normalize -> residual add.
// Wave-level reduction over 32 lanes via shfl_xor (wave32).
// ---------------------------------------------------------------------------
__global__ __launch_bounds__(256)
void node_epilogue(const float* __restrict__ xin, const float* __restrict__ agg,
                   const float* __restrict__ gamma, const float* __restrict__ beta,
                   float* __restrict__ hout)
{
    const int lane = threadIdx.x & 31;
    const int wave = threadIdx.x >> 5;
    const int n    = blockIdx.x * 8 + wave;     // 40000 % 8 == 0
    const int co   = lane * 4;

    float4 o  = *(const float4*)(agg   + (size_t)n * D + co);
    float4 g  = *(const float4*)(gamma + co);
    float4 bt = *(const float4*)(beta  + co);

    o.x = fmaxf(o.x * g.x + bt.x, 0.0f);
    o.y = fmaxf(o.y * g.y + bt.y, 0.0f);
    o.z = fmaxf(o.z * g.z + bt.z, 0.0f);
    o.w = fmaxf(o.w * g.w + bt.w, 0.0f);

    float ss = o.x * o.x + o.y * o.y + o.z * o.z + o.w * o.w;
    #pragma unroll
    for (int off = 16; off > 0; off >>= 1)
        ss += __shfl_xor(ss, off, 32);
    const float inv = 1.0f / fmaxf(sqrtf(ss), EPS_NORM);

    float4 xi = *(const float4*)(xin + (size_t)n * D + co);
    float4 h;
    h.x = xi.x + o.x * inv;
    h.y = xi.y + o.y * inv;
    h.z = xi.z + o.z * inv;
    h.w = xi.w + o.w * inv;
    *(float4*)(hout + (size_t)n * D + co) = h;   // same-row in-place safe
}

// ---------------------------------------------------------------------------
// Final: out = l2norm(h) per row.
// ---------------------------------------------------------------------------
__global__ __launch_bounds__(256)
void final_l2norm(const float* __restrict__ H, float* __restrict__ out)
{
    const int lane = threadIdx.x & 31;
    const int wave = threadIdx.x >> 5;
    const int n    = blockIdx.x * 8 + wave;
    const int co   = lane * 4;

    float4 h = *(const float4*)(H + (size_t)n * D + co);
    float ss = h.x * h.x + h.y * h.y + h.z * h.z + h.w * h.w;
    #pragma unroll
    for (int off = 16; off > 0; off >>= 1)
        ss += __shfl_xor(ss, off, 32);
    const float inv = 1.0f / fmaxf(sqrtf(ss), EPS_NORM);

    float4 o;
    o.x = h.x * inv; o.y = h.y * inv; o.z = h.z * inv; o.w = h.w * inv;
    *(float4*)(out + (size_t)n * D + co) = o;
}

// ---------------------------------------------------------------------------
extern "C" void kernel_launch(void* const* d_in, const int* in_sizes, int n_in,
                              void* d_out, int out_size, void* d_ws, size_t ws_size,
                              hipStream_t stream)
{
    const float* x     = (const float*)d_in[0];
    const int*   edges = (const int*)  d_in[1];   // [2, E]
    const float* Wk    = (const float*)d_in[2];
    const float* bk    = (const float*)d_in[3];
    const float* Wq    = (const float*)d_in[4];
    const float* bq    = (const float*)d_in[5];
    const float* Wv    = (const float*)d_in[6];
    const float* bv    = (const float*)d_in[7];
    const float* Ws    = (const float*)d_in[8];
    const float* gamma = (const float*)d_in[9];
    const float* beta  = (const float*)d_in[10];
    float*       out   = (float*)d_out;

    const int* src = edges;             // edge_index[0] = message source j
    const int* dst = edges + N_EDGES;   // edge_index[1] = message target i

    const size_t NS = (size_t)N_NODES * D;          // 5.12M floats / buffer
    float* Kb  = (float*)d_ws;
    float* Qb  = Kb  + NS;
    float* Vb  = Qb  + NS;
    float* Agg = Vb  + NS;
    float* H   = Agg + NS;                          // total 102.4 MB < ws_size

    const dim3 gemm_grid(N_NODES / 16);             // 2500 blocks
    const dim3 edge_grid(N_EDGES / 8);              // 80000 blocks (8 waves ea)
    const dim3 node_grid(N_NODES / 8);              // 5000 blocks
    const dim3 blk(256);

    for (int l = 0; l < LYR; ++l) {
        const float* xin = (l == 0) ? x : H;
        const size_t wo = (size_t)l * D * D;
        const size_t bo = (size_t)l * D;

        gemm_xt_wmma<<<gemm_grid, blk, 0, stream>>>(xin, Wk + wo, bk + bo, Kb);
        gemm_xt_wmma<<<gemm_grid, blk, 0, stream>>>(xin, Wq + wo, bq + bo, Qb);
        gemm_xt_wmma<<<gemm_grid, blk, 0, stream>>>(xin, Wv + wo, bv + bo, Vb);
        // Skip-connection GEMM seeds Agg (fully overwritten -> no memset),
        // edge messages then accumulate on top via L2 atomics.
        gemm_xt_wmma<<<gemm_grid, blk, 0, stream>>>(xin, Ws + wo, nullptr, Agg);

        edge_msg<<<edge_grid, blk, 0, stream>>>(src, dst, Kb, Qb, Vb, Agg);

        node_epilogue<<<node_grid, blk, 0, stream>>>(xin, Agg, gamma + bo, beta + bo, H);
    }

    final_l2norm<<<node_grid, blk, 0, stream>>>(H, out);
}